// NTLBGAttention_32882269618910
// MI455X (gfx1250) — compile-verified
//
#include <hip/hip_runtime.h>
#include <hip/hip_bf16.h>

#define D     768
#define RANK  64
#define NB    8
#define T     4096
#define KREP  6
#define MCAND 12
#define DR    (D * RANK)   // 49152
#define FR    32           // frames per block in dist kernel
#define KC    128          // K-chunk
#define WROW  132          // padded LDS row (conflict-avoiding stride, mod 64 == 4)

typedef float v2f __attribute__((ext_vector_type(2)));
typedef float v8f __attribute__((ext_vector_type(8)));

// ---------------- q = query @ Wq + bq ----------------
__global__ __launch_bounds__(256) void k_qproj(const float* __restrict__ query,
                                               const float* __restrict__ Wq,
                                               const float* __restrict__ bq,
                                               float* __restrict__ q) {
  __shared__ float qs[D];
  int b = blockIdx.x, t = threadIdx.x;
  for (int k = t; k < D; k += 256) qs[k] = query[b * D + k];
  __syncthreads();
  for (int j = t; j < D; j += 256) {
    float acc = bq[j];
    for (int k = 0; k < D; ++k) acc = fmaf(qs[k], Wq[k * D + j], acc);
    q[b * D + j] = acc;
  }
}

// ------- mu = relu(LN(q@Wm1+bm1))@Wm2+bm2 ; dis = 1/sqrt(softplus(q@Wd+bd)+2e-6) -------
__global__ __launch_bounds__(256) void k_mu_diag(const float* __restrict__ q,
                                                 const float* __restrict__ Wm1,
                                                 const float* __restrict__ bm1,
                                                 const float* __restrict__ lng,
                                                 const float* __restrict__ lnb,
                                                 const float* __restrict__ Wm2,
                                                 const float* __restrict__ bm2,
                                                 const float* __restrict__ Wd,
                                                 const float* __restrict__ bd,
                                                 float* __restrict__ mu,
                                                 float* __restrict__ dis) {
  __shared__ float qs[D];
  __shared__ float hs[D];
  __shared__ float red[256];
  __shared__ float red2[256];
  int b = blockIdx.x, t = threadIdx.x;
  for (int k = t; k < D; k += 256) qs[k] = q[b * D + k];
  __syncthreads();
  float ls = 0.f, ls2 = 0.f;
  for (int j = t; j < D; j += 256) {
    float a1 = bm1[j], a2 = bd[j];
    for (int k = 0; k < D; ++k) {
      float qk = qs[k];
      a1 = fmaf(qk, Wm1[k * D + j], a1);
      a2 = fmaf(qk, Wd[k * D + j], a2);
    }
    hs[j] = a1;
    ls += a1;
    ls2 += a1 * a1;
    float sp = (a2 > 20.f) ? a2 : log1pf(expf(a2));
    dis[b * D + j] = rsqrtf(sp + 2e-6f);   // softplus + EPS (diag) + EPS (chol jitter)
  }
  red[t] = ls;
  red2[t] = ls2;
  __syncthreads();
  for (int s = 128; s > 0; s >>= 1) {
    if (t < s) { red[t] += red[t + s]; red2[t] += red2[t + s]; }
    __syncthreads();
  }
  float mean = red[0] * (1.f / D);
  float var  = red2[0] * (1.f / D) - mean * mean;
  float inv  = rsqrtf(var + 1e-5f);
  for (int j = t; j < D; j += 256) {
    float hn = (hs[j] - mean) * inv * lng[j] + lnb[j];
    hs[j] = fmaxf(hn, 0.f);
  }
  __syncthreads();
  for (int j = t; j < D; j += 256) {
    float acc = bm2[j];
    for (int k = 0; k < D; ++k) acc = fmaf(hs[k], Wm2[k * D + j], acc);
    mu[b * D + j] = acc;
  }
}

// --------- Vt[b][r][i] = (q@Wu+bu)[b][i][r] * dis[b][i]   (streams 151MB of Wu once) ---------
__global__ __launch_bounds__(256) void k_uproj(const float* __restrict__ q,
                                               const float* __restrict__ Wu,
                                               const float* __restrict__ bu,
                                               const float* __restrict__ dis,
                                               float* __restrict__ Vt) {
  __shared__ float qs[NB * D];   // 24KB
  int t = threadIdx.x;
  for (int k = t; k < NB * D; k += 256) qs[k] = q[k];
  __syncthreads();
  int j = blockIdx.x * 256 + t;           // output column in [0, 49152)
  float acc[NB];
#pragma unroll
  for (int b = 0; b < NB; ++b) acc[b] = 0.f;
  for (int k = 0; k < D; ++k) {
    float wv = Wu[(size_t)k * DR + j];    // coalesced stream of Wu
#pragma unroll
    for (int b = 0; b < NB; ++b) acc[b] = fmaf(qs[b * D + k], wv, acc[b]);
  }
  int i = j >> 6, r = j & 63;
  float bias = bu[j];
#pragma unroll
  for (int b = 0; b < NB; ++b)
    Vt[(size_t)b * DR + r * D + i] = (acc[b] + bias) * dis[b * D + i];
}

// --------- M = I + Vt Vt^T (64x64), Minv = M^{-1} via Gauss-Jordan (SPD, no pivoting) ---------
__global__ __launch_bounds__(256) void k_minv(const float* __restrict__ Vt,
                                              float* __restrict__ Minv) {
  __shared__ float A[RANK][RANK + 1];
  __shared__ float Iv[RANK][RANK + 1];
  __shared__ float colk[RANK];
  __shared__ float pivInvS;
  int b = blockIdx.x, t = threadIdx.x;
  const float* V = Vt + (size_t)b * DR;
  for (int e = t; e < RANK * RANK; e += 256) {
    int r = e >> 6, s = e & 63;
    float acc = (r == s) ? 1.f : 0.f;
    const float* vr = V + r * D;
    const float* vs = V + s * D;
    for (int k = 0; k < D; ++k) acc = fmaf(vr[k], vs[k], acc);
    A[r][s] = acc;
    Iv[r][s] = (r == s) ? 1.f : 0.f;
  }
  __syncthreads();
  for (int k = 0; k < RANK; ++k) {
    if (t == 0) pivInvS = 1.f / A[k][k];
    __syncthreads();
    float pinv = pivInvS;
    if (t < RANK) { A[k][t] *= pinv; Iv[k][t] *= pinv; }
    __syncthreads();
    if (t < RANK) colk[t] = A[t][k];
    __syncthreads();
    for (int e = t; e < RANK * RANK; e += 256) {
      int i = e >> 6, jx = e & 63;
      if (i != k) {
        float f = colk[i];
        A[i][jx]  = fmaf(-f, A[k][jx], A[i][jx]);
        Iv[i][jx] = fmaf(-f, Iv[k][jx], Iv[i][jx]);
      }
    }
    __syncthreads();
  }
  for (int e = t; e < RANK * RANK; e += 256)
    Minv[(size_t)b * RANK * RANK + e] = Iv[e >> 6][e & 63];
}

// --------- Mahalanobis distances for all frames via Woodbury + f32 WMMA ---------
// dist[n] = |w_n|^2 - g_n^T Minv g_n,  w = D^{-1/2}(x-mu),  g = Vt w  (G via V_WMMA_F32_16X16X4_F32)
__global__ __launch_bounds__(256) void k_dist(const float* __restrict__ feats,
                                              const float* __restrict__ mu,
                                              const float* __restrict__ dis,
                                              const float* __restrict__ Vt,
                                              const float* __restrict__ Minv,
                                              float* __restrict__ dist) {
  __shared__ float wl[FR][WROW];       // centered/scaled frame tile (16.9KB)
  __shared__ float gl[RANK][FR + 1];   // G tile (8.4KB)
  __shared__ float mv[RANK][RANK];     // Minv (16KB)
  __shared__ float sumsq[FR];
  __shared__ float quad[FR];
  int t = threadIdx.x;
  int b = blockIdx.y;
  int n0 = blockIdx.x * FR;

  for (int e = t; e < RANK * RANK; e += 256) mv[e >> 6][e & 63] = Minv[(size_t)b * RANK * RANK + e];
  if (t < FR) { sumsq[t] = 0.f; quad[t] = 0.f; }

  int f  = t >> 3;               // frame within tile: 0..31
  int ks = (t & 7) * 16;         // k strip within chunk
  const float* frow   = feats + ((size_t)(b * T + n0 + f)) * D;
  const float* murow  = mu  + b * D;
  const float* disrow = dis + b * D;

  int lane = t & 31;
  int wv   = t >> 5;             // wave 0..7
  int mT   = wv & 3;             // M tile (rank block)
  int nS   = wv >> 2;            // N subtile (frame block)
  int bcol0 = (lane >> 4) << 1;  // lanes 0-15 -> K{0,1}, lanes 16-31 -> K{2,3}
  const float* arow = Vt + (size_t)b * DR + (mT * 16 + (lane & 15)) * D + bcol0;

  v8f cacc = {0.f, 0.f, 0.f, 0.f, 0.f, 0.f, 0.f, 0.f};
  float sq = 0.f;

  for (int kc = 0; kc < D; kc += KC) {
    __syncthreads();
    // stage w = (x - mu) * dinv_sqrt into LDS, accumulate |w|^2 partials
#pragma unroll
    for (int e = 0; e < 16; ++e) {
      int k = kc + ks + e;
      float x = frow[k];
      float wvv = (x - murow[k]) * disrow[k];
      wl[f][ks + e] = wvv;
      sq = fmaf(wvv, wvv, sq);
    }
    __syncthreads();
    const float* ap = arow + kc;
#pragma unroll 8
    for (int s = 0; s < KC / 4; ++s) {
      v2f a = *(const v2f*)(ap + s * 4);
      v2f bb = *(const v2f*)&wl[nS * 16 + (lane & 15)][s * 4 + bcol0];
      cacc = __builtin_amdgcn_wmma_f32_16x16x4_f32(
          /*neg_a=*/false, a, /*neg_b=*/false, bb,
          /*c_mod=*/(short)0, cacc, /*reuse_a=*/false, /*reuse_b=*/false);
    }
  }
  atomicAdd(&sumsq[f], sq);

  // scatter C tile (VGPR v, lane L) -> G[mT*16 + v + (L/16)*8][nS*16 + L%16]
  {
    int gn = nS * 16 + (lane & 15);
    int mb = mT * 16 + ((lane >> 4) << 3);
#pragma unroll
    for (int v = 0; v < 8; ++v) gl[mb + v][gn] = cacc[v];
  }
  __syncthreads();

  // quadratic form g^T Minv g, 8 row-groups per frame reduced via LDS atomics
  {
    int ff = t & 31;
    int rg = t >> 5;
    float acc = 0.f;
    for (int rr = 0; rr < 8; ++rr) {
      int r = rg * 8 + rr;
      float inner = 0.f;
      for (int s = 0; s < RANK; ++s) inner = fmaf(mv[r][s], gl[s][ff], inner);
      acc = fmaf(gl[r][ff], inner, acc);
    }
    atomicAdd(&quad[ff], acc);
  }
  __syncthreads();
  if (t < FR) dist[(size_t)b * T + n0 + t] = sumsq[t] - quad[t];
}

// --------- NTLBG selection: rank-2048 target, top-12 by |d-target|, greedy diversify ---------
__global__ __launch_bounds__(256) void k_select(const float* __restrict__ dist,
                                                const float* __restrict__ feats,
                                                float* __restrict__ out_idx,
                                                float* __restrict__ out_rep,
                                                float* __restrict__ repd,
                                                int* __restrict__ selg) {
  __shared__ float ds[T];        // 16KB
  __shared__ float targetS;
  __shared__ float rv[256];
  __shared__ int   ri[256];
  __shared__ int   cand[MCAND];
  __shared__ int   selS[KREP];
  int b = blockIdx.x, t = threadIdx.x;
  for (int e = t; e < T; e += 256) ds[e] = dist[(size_t)b * T + e];
  __syncthreads();
  // element of rank T/2 (== sorted_d[T//2]); total order with index tie-break
  for (int e = t; e < T; e += 256) {
    float de = ds[e];
    int cnt = 0;
    for (int j = 0; j < T; ++j) {
      float dj = ds[j];
      cnt += (dj < de || (dj == de && j < e)) ? 1 : 0;
    }
    if (cnt == T / 2) targetS = de;
  }
  __syncthreads();
  float tg = targetS;
  // iterative top-12 smallest |d - target| (lowest index on ties, matching lax.top_k set)
  for (int it = 0; it < MCAND; ++it) {
    float bd = 3.4e38f;
    int bi = T;
    for (int e = t; e < T; e += 256) {
      bool tkn = false;
      for (int p = 0; p < it; ++p) tkn |= (cand[p] == e);
      if (!tkn) {
        float df = fabsf(ds[e] - tg);
        if (df < bd || (df == bd && e < bi)) { bd = df; bi = e; }
      }
    }
    rv[t] = bd; ri[t] = bi;
    __syncthreads();
    for (int s = 128; s > 0; s >>= 1) {
      if (t < s) {
        if (rv[t + s] < rv[t] || (rv[t + s] == rv[t] && ri[t + s] < ri[t])) {
          rv[t] = rv[t + s]; ri[t] = ri[t + s];
        }
      }
      __syncthreads();
    }
    if (t == 0) cand[it] = ri[0];
    __syncthreads();
  }
  if (t == 0) {
    // sort candidate indices ascending (temporal order)
    for (int i = 1; i < MCAND; ++i) {
      int v = cand[i], j = i - 1;
      while (j >= 0 && cand[j] > v) { cand[j + 1] = cand[j]; --j; }
      cand[j + 1] = v;
    }
    int sel[KREP];
    bool taken[MCAND];
    for (int i = 0; i < MCAND; ++i) taken[i] = false;
    for (int i = 0; i < KREP; ++i) sel[i] = cand[0];
    taken[0] = true;
    for (int i = 1; i < KREP; ++i) {
      int bestScore = -2, bestJ = 0;
      for (int j = 0; j < MCAND; ++j) {
        int score;
        if (taken[j]) score = -1;
        else {
          int mind = 0x7fffffff;
          for (int sl = 0; sl < KREP; ++sl) {
            int dd = cand[j] - sel[sl];
            if (dd < 0) dd = -dd;
            if (dd < mind) mind = dd;
          }
          score = mind;
        }
        if (score > bestScore) { bestScore = score; bestJ = j; }  // first max
      }
      sel[i] = cand[bestJ];
      taken[bestJ] = true;
    }
    for (int i = 0; i < KREP; ++i) {
      selS[i] = sel[i];
      out_idx[b * KREP + i] = (float)sel[i];
      repd[b * KREP + i] = ds[sel[i]];
      selg[b * KREP + i] = sel[i];
    }
  }
  __syncthreads();
  for (int e = t; e < KREP * D; e += 256) {
    int i = e / D, k = e % D;
    out_rep[((size_t)b * KREP + i) * D + k] = feats[((size_t)b * T + selS[i]) * D + k];
  }
}

// --------- weights: 0.7*softmax(-repdist/T) + 0.3*softmax(sigmoid(MLP(rep))) ---------
__global__ __launch_bounds__(256) void k_weights(const float* __restrict__ rep,
                                                 const float* __restrict__ Wp1,
                                                 const float* __restrict__ bp1,
                                                 const float* __restrict__ Wp2,
                                                 const float* __restrict__ bp2,
                                                 const float* __restrict__ repd,
                                                 float* __restrict__ out_w) {
  __shared__ float hp[KREP][D / 2];   // 9.2KB
  __shared__ float neural[KREP];
  int b = blockIdx.x, t = threadIdx.x;
  const float* reprow = rep + (size_t)b * KREP * D;
  for (int e = t; e < KREP * (D / 2); e += 256) {
    int i = e / (D / 2), j = e % (D / 2);
    float acc = bp1[j];
    const float* rr = reprow + i * D;
    for (int k = 0; k < D; ++k) acc = fmaf(rr[k], Wp1[k * (D / 2) + j], acc);
    hp[i][j] = fmaxf(acc, 0.f);
  }
  __syncthreads();
  if (t < KREP) {
    float acc = bp2[0];
    for (int k = 0; k < D / 2; ++k) acc = fmaf(hp[t][k], Wp2[k], acc);
    neural[t] = 1.f / (1.f + expf(-acc));
  }
  __syncthreads();
  if (t == 0) {
    float dw[KREP], nw[KREP];
    float mx = -3.4e38f;
    for (int i = 0; i < KREP; ++i) {
      float v = -repd[b * KREP + i] * (1.f / 0.07f);
      dw[i] = v;
      if (v > mx) mx = v;
    }
    float s1 = 0.f;
    for (int i = 0; i < KREP; ++i) { dw[i] = expf(dw[i] - mx); s1 += dw[i]; }
    float mx2 = -3.4e38f;
    for (int i = 0; i < KREP; ++i) if (neural[i] > mx2) mx2 = neural[i];
    float s2 = 0.f;
    for (int i = 0; i < KREP; ++i) { nw[i] = expf(neural[i] - mx2); s2 += nw[i]; }
    for (int i = 0; i < KREP; ++i)
      out_w[b * KREP + i] = 0.7f * dw[i] / s1 + 0.3f * nw[i] / s2;
  }
}

extern "C" void kernel_launch(void* const* d_in, const int* in_sizes, int n_in,
                              void* d_out, int out_size, void* d_ws, size_t ws_size,
                              hipStream_t stream) {
  (void)in_sizes; (void)n_in; (void)out_size; (void)ws_size;
  const float* feats = (const float*)d_in[0];
  const float* query = (const float*)d_in[1];
  const float* Wq  = (const float*)d_in[2];
  const float* bq  = (const float*)d_in[3];
  const float* Wm1 = (const float*)d_in[4];
  const float* bm1 = (const float*)d_in[5];
  const float* lng = (const float*)d_in[6];
  const float* lnb = (const float*)d_in[7];
  const float* Wm2 = (const float*)d_in[8];
  const float* bm2 = (const float*)d_in[9];
  const float* Wu  = (const float*)d_in[10];
  const float* bu  = (const float*)d_in[11];
  const float* Wd  = (const float*)d_in[12];
  const float* bd  = (const float*)d_in[13];
  const float* Wp1 = (const float*)d_in[14];
  const float* bp1 = (const float*)d_in[15];
  const float* Wp2 = (const float*)d_in[16];
  const float* bp2 = (const float*)d_in[17];

  float* ws   = (float*)d_ws;
  float* q    = ws;                                 // 8*768
  float* mu   = q + NB * D;                         // 8*768
  float* dis  = mu + NB * D;                        // 8*768
  float* Vt   = dis + NB * D;                       // 8*64*768
  float* Minv = Vt + (size_t)NB * DR;               // 8*64*64
  float* dstb = Minv + NB * RANK * RANK;            // 8*4096
  float* repd = dstb + (size_t)NB * T;              // 48
  int*   selg = (int*)(repd + 64);                  // 48

  float* out_idx = (float*)d_out;                   // 48
  float* out_rep = out_idx + NB * KREP;             // 8*6*768
  float* out_w   = out_rep + (size_t)NB * KREP * D; // 48

  k_qproj<<<NB, 256, 0, stream>>>(query, Wq, bq, q);
  k_mu_diag<<<NB, 256, 0, stream>>>(q, Wm1, bm1, lng, lnb, Wm2, bm2, Wd, bd, mu, dis);
  k_uproj<<<DR / 256, 256, 0, stream>>>(q, Wu, bu, dis, Vt);
  k_minv<<<NB, 256, 0, stream>>>(Vt, Minv);
  dim3 gdist(T / FR, NB);
  k_dist<<<gdist, 256, 0, stream>>>(feats, mu, dis, Vt, Minv, dstb);
  k_select<<<NB, 256, 0, stream>>>(dstb, feats, out_idx, out_rep, repd, selg);
  k_weights<<<NB, 256, 0, stream>>>(out_rep, Wp1, bp1, Wp2, bp2, repd, out_w);
}